// L1Writer_29661044146657
// MI455X (gfx1250) — compile-verified
//
#include <hip/hip_runtime.h>

typedef float v2f __attribute__((ext_vector_type(2)));
typedef float v8f __attribute__((ext_vector_type(8)));

#define NHEAD   16
#define DK      64
#define DV      64
#define NTOT    16384        // B*S = 4*4096
#define CHUNKS  32           // split-K factor -> 512 workgroups
#define NB      128          // rows staged in LDS per block
#define LSTRIDE 80           // 64 + 16 pad: row r bank = (16r + col) % 64 -> conflict-free WMMA gathers
#define DECAY   0.95f

// out = DECAY * memory  (atomically accumulated into afterwards)
__global__ __launch_bounds__(512) void l1w_init(const float* __restrict__ mem,
                                                float* __restrict__ out) {
    int i = blockIdx.x * 512 + threadIdx.x;
    if (i < NHEAD * DK * DV) out[i] = DECAY * mem[i];
}

// Per workgroup: one head, one 512-row slice of the reduction.
// 16 waves each own one 16x16 tile of the 64x64 per-head output.
// Register double-buffering: global loads for block i+1 are issued right
// after the staging barrier of block i, so HBM latency hides behind the
// 32-WMMA compute phase.
__global__ __launch_bounds__(512) void l1w_wmma(const float* __restrict__ keys,
                                                const float* __restrict__ vals,
                                                const float* __restrict__ rho,
                                                float* __restrict__ out) {
    __shared__ float kl[NB * LSTRIDE];
    __shared__ float vl[NB * LSTRIDE];

    const int head  = blockIdx.x / CHUNKS;
    const int chunk = blockIdx.x % CHUNKS;
    const int rows_per_wg = NTOT / CHUNKS;   // 512
    const int row0  = chunk * rows_per_wg;

    const int tid  = threadIdx.x;
    const int wave = tid >> 5;
    const int lane = tid & 31;
    const int hlf  = lane >> 4;              // 0: lanes 0-15, 1: lanes 16-31
    const int l15  = lane & 15;
    const int tm   = (wave >> 2) * 16;       // dk tile offset
    const int tn   = (wave & 3)  * 16;       // dv tile offset

    // Staging assignment: 512 threads * 4 iters, one float4 per (thread,iter).
    const int r_base = tid >> 4;             // row within block, +32 per iter
    const int c_stg  = (tid & 15) << 2;      // float4-aligned column

    v8f acc = {};

    float4 kreg[4], vreg[4];
    float  wreg[4];

    // Prologue: fetch block 0.
    #pragma unroll
    for (int it = 0; it < 4; ++it) {
        int g = row0 + it * 32 + r_base;
        kreg[it] = *(const float4*)(keys + ((size_t)g * NHEAD + head) * DK + c_stg);
        vreg[it] = *(const float4*)(vals + ((size_t)g * NHEAD + head) * DV + c_stg);
        wreg[it] = rho[g];
    }

    for (int blk = 0; blk < rows_per_wg; blk += NB) {
        // Drain registers into LDS (fold rho into K here).
        #pragma unroll
        for (int it = 0; it < 4; ++it) {
            int r = it * 32 + r_base;
            float* kd = &kl[r * LSTRIDE + c_stg];
            float* vd = &vl[r * LSTRIDE + c_stg];
            const float w = wreg[it];
            kd[0] = kreg[it].x * w; kd[1] = kreg[it].y * w;
            kd[2] = kreg[it].z * w; kd[3] = kreg[it].w * w;
            vd[0] = vreg[it].x;     vd[1] = vreg[it].y;
            vd[2] = vreg[it].z;     vd[3] = vreg[it].w;
        }
        __syncthreads();

        // Issue next block's global loads before computing (latency hiding).
        if (blk + NB < rows_per_wg) {
            #pragma unroll
            for (int it = 0; it < 4; ++it) {
                int g = row0 + blk + NB + it * 32 + r_base;
                kreg[it] = *(const float4*)(keys + ((size_t)g * NHEAD + head) * DK + c_stg);
                vreg[it] = *(const float4*)(vals + ((size_t)g * NHEAD + head) * DV + c_stg);
                wreg[it] = rho[g];
            }
        }

        // A[M=dk][kk=s] from weighted-K, B[kk=s][N=dv] from V.
        // 16x4 f32 A layout: v0 = kk {0 | 2}, v1 = kk {1 | 3} across lane halves.
        #pragma unroll 4
        for (int s = 0; s < NB; s += 4) {
            int ra = s + hlf * 2;
            v2f a, b;
            a.x = kl[ ra      * LSTRIDE + tm + l15];
            a.y = kl[(ra + 1) * LSTRIDE + tm + l15];
            b.x = vl[ ra      * LSTRIDE + tn + l15];
            b.y = vl[(ra + 1) * LSTRIDE + tn + l15];
            acc = __builtin_amdgcn_wmma_f32_16x16x4_f32(
                /*neg_a=*/false, a, /*neg_b=*/false, b,
                /*c_mod=*/(short)0, acc, /*reuse_a=*/false, /*reuse_b=*/false);
        }
        __syncthreads();
    }

    // C/D layout: VGPR i holds M = i (lanes 0-15) / M = i+8 (lanes 16-31), N = lane&15.
    #pragma unroll
    for (int i = 0; i < 8; ++i) {
        int dk = tm + i + hlf * 8;
        int dv = tn + l15;
        atomicAdd(&out[((size_t)head * DK + dk) * DV + dv], acc[i]);
    }
}

extern "C" void kernel_launch(void* const* d_in, const int* in_sizes, int n_in,
                              void* d_out, int out_size, void* d_ws, size_t ws_size,
                              hipStream_t stream) {
    const float* mem  = (const float*)d_in[0];   // (H, Dk, Dv)
    const float* keys = (const float*)d_in[1];   // (B, S, H, Dk)
    const float* vals = (const float*)d_in[2];   // (B, S, H, Dv)
    const float* rho  = (const float*)d_in[3];   // (B, S)
    float* out = (float*)d_out;                  // (H, Dk, Dv)

    l1w_init<<<(NHEAD * DK * DV + 511) / 512, 512, 0, stream>>>(mem, out);
    l1w_wmma<<<NHEAD * CHUNKS, 512, 0, stream>>>(keys, vals, rho, out);
}